// PromptLearner_Conditional_9259949490298
// MI455X (gfx1250) — compile-verified
//
#include <hip/hip_runtime.h>
#include <hip/hip_bf16.h>

typedef float v2f __attribute__((ext_vector_type(2)));
typedef float v8f __attribute__((ext_vector_type(8)));
typedef float f4  __attribute__((ext_vector_type(4)));

#define N_PAIR 32
#define N_CLS  92     // NUM_BASE
#define N_CTX  8
#define DMODEL 768
#define MAX_L  40
#define L_SUF  31     // MAX_L - 1 - N_CTX

// ---------------------------------------------------------------------------
// Kernel 1: h = relu([enti[s],enti[o]] @ W1 + b1)   (32x512)@(512x256)->(32x256)
// One block, 8 waves, 32 WMMA tiles (2 M-tiles x 16 N-tiles), 4 tiles/wave.
// ---------------------------------------------------------------------------
__global__ __launch_bounds__(256) void k_gemm1(const int* __restrict__ so_ids,
                                               const float* __restrict__ enti,
                                               const float* __restrict__ W1,
                                               const float* __restrict__ b1,
                                               float* __restrict__ h) {
    const int tid  = threadIdx.x;
    const int wave = tid >> 5;
    const int lane = tid & 31;
    const int lmod = lane & 15;   // 0..15
    const int lhi  = lane >> 4;   // 0 or 1

    for (int i = 0; i < 4; ++i) {
        const int tile = wave * 4 + i;      // 0..31
        const int mt = tile >> 4;           // 0..1
        const int nt = tile & 15;           // 0..15
        const int p   = mt * 16 + lmod;     // A row this lane supplies
        const int sid = so_ids[2 * p + 0];
        const int oid = so_ids[2 * p + 1];
        const int col = nt * 16 + lmod;     // B/D column this lane supplies

        v8f acc = {};
        for (int k0 = 0; k0 < 512; k0 += 4) {
            const int kb = k0 + 2 * lhi;
            v2f a, b;
#pragma unroll
            for (int r = 0; r < 2; ++r) {
                const int k = kb + r;
                // gather so[p][k] = k<256 ? enti[sid][k] : enti[oid][k-256]
                const float vs = enti[sid * 256 + (k & 255)];
                const float vo = enti[oid * 256 + (k & 255)];
                a[r] = (k < 256) ? vs : vo;          // branchless (v_cndmask)
                b[r] = W1[k * 256 + col];
            }
            acc = __builtin_amdgcn_wmma_f32_16x16x4_f32(
                false, a, false, b, (short)0, acc, false, false);
        }
        const float bias = b1[col];
#pragma unroll
        for (int r = 0; r < 8; ++r) {
            const int row = mt * 16 + r + 8 * lhi;   // D layout
            const float v = acc[r] + bias;
            h[row * 256 + col] = v > 0.0f ? v : 0.0f;
        }
    }
}

// ---------------------------------------------------------------------------
// Kernel 2: so_e = h @ W2   (32x256)@(256x1536)->(32x1536)
// 24 blocks x 8 waves = 192 tiles (2 x 96), one tile per wave.
// ---------------------------------------------------------------------------
__global__ __launch_bounds__(256) void k_gemm2(const float* __restrict__ h,
                                               const float* __restrict__ W2,
                                               float* __restrict__ soe) {
    const int tid  = threadIdx.x;
    const int wave = tid >> 5;
    const int lane = tid & 31;
    const int lmod = lane & 15;
    const int lhi  = lane >> 4;

    const int tile = blockIdx.x * 8 + wave;  // 0..191
    const int mt = tile / 96;                // 0..1
    const int nt = tile % 96;                // 0..95
    const int arow = mt * 16 + lmod;
    const int col  = nt * 16 + lmod;

    v8f acc = {};
    for (int k0 = 0; k0 < 256; k0 += 4) {
        const int kb = k0 + 2 * lhi;
        v2f a, b;
        a[0] = h[arow * 256 + kb];
        a[1] = h[arow * 256 + kb + 1];
        b[0] = W2[kb * 1536 + col];
        b[1] = W2[(kb + 1) * 1536 + col];
        acc = __builtin_amdgcn_wmma_f32_16x16x4_f32(
            false, a, false, b, (short)0, acc, false, false);
    }
#pragma unroll
    for (int r = 0; r < 8; ++r) {
        const int row = mt * 16 + r + 8 * lhi;
        soe[row * 1536 + col] = acc[r];
    }
}

// ---------------------------------------------------------------------------
// Kernel 3: attention. 64 blocks (32 subj + 32 obj queries), 256 threads.
// att = softmax(q @ meta^T / sqrt(D)) @ meta
// ---------------------------------------------------------------------------
__global__ __launch_bounds__(256) void k_attn(const float* __restrict__ soe,
                                              const float* __restrict__ meta,
                                              float* __restrict__ satt,
                                              float* __restrict__ oatt) {
    __shared__ float red[8][256];
    __shared__ float wts[8];
    const int tid  = threadIdx.x;
    const int q    = blockIdx.x;     // 0..63
    const int p    = q & 31;
    const int is_o = q >> 5;
    const float* qv = soe + p * 1536 + is_o * DMODEL;

    float part[8] = {0.f,0.f,0.f,0.f,0.f,0.f,0.f,0.f};
    for (int d = tid; d < DMODEL; d += 256) {
        const float x = qv[d];
#pragma unroll
        for (int j = 0; j < 8; ++j) part[j] += x * meta[j * DMODEL + d];
    }
#pragma unroll
    for (int j = 0; j < 8; ++j) red[j][tid] = part[j];
    __syncthreads();
    for (int s = 128; s > 0; s >>= 1) {
        if (tid < s) {
#pragma unroll
            for (int j = 0; j < 8; ++j) red[j][tid] += red[j][tid + s];
        }
        __syncthreads();
    }
    if (tid == 0) {
        const float scale = 0.03608439182435161f;  // 1/sqrt(768)
        float l[8], m = -1e30f;
#pragma unroll
        for (int j = 0; j < 8; ++j) { l[j] = red[j][0] * scale; m = l[j] > m ? l[j] : m; }
        float s = 0.f;
#pragma unroll
        for (int j = 0; j < 8; ++j) { l[j] = expf(l[j] - m); s += l[j]; }
        const float inv = 1.0f / s;
#pragma unroll
        for (int j = 0; j < 8; ++j) wts[j] = l[j] * inv;
    }
    __syncthreads();
    float w[8];
#pragma unroll
    for (int j = 0; j < 8; ++j) w[j] = wts[j];
    float* out = (is_o ? oatt : satt) + p * DMODEL;
    for (int d = tid; d < DMODEL; d += 256) {
        float acc = 0.f;
#pragma unroll
        for (int j = 0; j < 8; ++j) acc += w[j] * meta[j * DMODEL + d];
        out[d] = acc;
    }
}

// ---------------------------------------------------------------------------
// Kernel 4: the bandwidth kernel. Builds both (2944,40,768) tensors with
// 128-bit coalesced non-temporal stores; linear float4 index == output offset.
// ---------------------------------------------------------------------------
__global__ __launch_bounds__(256) void k_build(const float* __restrict__ prefix,
                                               const float* __restrict__ suffix,
                                               const float* __restrict__ subj_ctx,
                                               const float* __restrict__ obj_ctx,
                                               const float* __restrict__ satt,
                                               const float* __restrict__ oatt,
                                               float* __restrict__ out) {
    const int  ROW4  = DMODEL / 4;                                  // 192
    const long long HALF4 = 1LL * N_PAIR * N_CLS * MAX_L * ROW4;    // 22,609,920
    const long long TOT4  = 2 * HALF4;
    const long long stride = (long long)gridDim.x * blockDim.x;

    for (long long v = (long long)blockIdx.x * blockDim.x + threadIdx.x;
         v < TOT4; v += stride) {
        long long e = v;
        const int which = (int)(e / HALF4);     // 0 = subj, 1 = obj
        e -= (long long)which * HALF4;
        const int pc = (int)(e / (MAX_L * ROW4));
        const int r  = (int)(e - (long long)pc * (MAX_L * ROW4));
        const int l  = r / ROW4;
        const int d4 = r - l * ROW4;
        const int p  = pc / N_CLS;
        const int c  = pc - p * N_CLS;
        const int d  = d4 * 4;

        f4 val;
        if (l == 0) {
            val = *(const f4*)(prefix + (c + 1) * DMODEL + d);
        } else if (l <= N_CTX) {
            const float* ctx = which ? obj_ctx : subj_ctx;
            const float* att = which ? oatt : satt;
            f4 a = *(const f4*)(ctx + (l - 1) * DMODEL + d);
            f4 b = *(const f4*)(att + p * DMODEL + d);
            val = a + b;
        } else {
            val = *(const f4*)(suffix +
                               ((long long)(c + 1) * L_SUF + (l - 1 - N_CTX)) * DMODEL + d);
        }
        __builtin_nontemporal_store(val, (f4*)(out + v * 4));
    }
}

// ---------------------------------------------------------------------------
// Kernel 5: mask = tile(token_mask[1..92], (32,1)) -> int32 tail of d_out
// ---------------------------------------------------------------------------
__global__ __launch_bounds__(256) void k_mask(const int* __restrict__ token_mask,
                                              int* __restrict__ mout) {
    const int TOT = N_PAIR * N_CLS * MAX_L;   // 117760
    const int idx = blockIdx.x * 256 + threadIdx.x;
    if (idx < TOT) {
        const int l  = idx % MAX_L;
        const int pc = idx / MAX_L;
        const int c  = pc % N_CLS;
        mout[idx] = token_mask[(c + 1) * MAX_L + l];
    }
}

// ---------------------------------------------------------------------------
extern "C" void kernel_launch(void* const* d_in, const int* in_sizes, int n_in,
                              void* d_out, int out_size, void* d_ws, size_t ws_size,
                              hipStream_t stream) {
    const int*   so_ids     = (const int*)d_in[0];
    const float* enti       = (const float*)d_in[1];
    const float* prefix     = (const float*)d_in[2];
    const float* suffix     = (const float*)d_in[3];
    const int*   token_mask = (const int*)d_in[4];
    const float* meta       = (const float*)d_in[5];
    const float* subj_ctx   = (const float*)d_in[6];
    const float* obj_ctx    = (const float*)d_in[7];
    const float* W1         = (const float*)d_in[8];
    const float* b1         = (const float*)d_in[9];
    const float* W2         = (const float*)d_in[10];

    float* ws   = (float*)d_ws;
    float* h    = ws;                       // 32*256
    float* soe  = h + 32 * 256;             // 32*1536
    float* satt = soe + 32 * 1536;          // 32*768
    float* oatt = satt + 32 * 768;          // 32*768

    float* out = (float*)d_out;
    const long long HALF = 1LL * N_PAIR * N_CLS * MAX_L * DMODEL;   // 90,439,680
    int* mout = (int*)(out + 2 * HALF);

    k_gemm1<<<1, 256, 0, stream>>>(so_ids, enti, W1, b1, h);
    k_gemm2<<<24, 256, 0, stream>>>(h, W2, soe);
    k_attn<<<64, 256, 0, stream>>>(soe, meta, satt, oatt);
    k_build<<<4096, 256, 0, stream>>>(prefix, suffix, subj_ctx, obj_ctx, satt, oatt, out);
    k_mask<<<(N_PAIR * N_CLS * MAX_L + 255) / 256, 256, 0, stream>>>(token_mask, mout);
}